// EEG_GAT_2095944040796
// MI455X (gfx1250) — compile-verified
//
#include <hip/hip_runtime.h>

// Problem constants (from reference)
#define B_   1024
#define C_   63
#define F_   250
#define H_   4
#define D_   250
#define HD_  1000
#define TOPK 8
#define NEG  0.2f

typedef float v2f __attribute__((ext_vector_type(2)));
typedef float v8f __attribute__((ext_vector_type(8)));

// CDNA5 fp32 tensor op: D(16x16) = A(16x4) * B(4x16) + C
__device__ __forceinline__ v8f wmma4(v2f a, v2f b, v8f c) {
  return __builtin_amdgcn_wmma_f32_16x16x4_f32(false, a, false, b, (short)0, c,
                                               false, false);
}

// ---------------------------------------------------------------------------
// Kernel 1: per-block partial gram of standardized rows.
// 64 blocks x 16 batches each; 16 waves -> 4x4 grid of 16x16 WMMA tiles
// covering the (padded) 64x64 gram. Deterministic: partials written per block.
// ---------------------------------------------------------------------------
__global__ __launch_bounds__(512) void k_adj(const float* __restrict__ x,
                                             float* __restrict__ padj) {
  __shared__ float xn[64][252];   // rows padded 63->64, K padded 250->252
  __shared__ float mu[64];
  __shared__ float rs[64];
  const int tid  = threadIdx.x;
  const int lane = tid & 31;
  const int wv   = tid >> 5;
  const int ti   = wv >> 2, tj = wv & 3;
  const int m16  = lane & 15;
  const int khi  = (lane >> 4) << 1;   // K sub-offset per ISA A-layout
  v8f acc = {};
  for (int bb = 0; bb < 16; ++bb) {
    const int b = blockIdx.x * 16 + bb;
    for (int idx = tid; idx < 64 * 252; idx += 512) {
      int r = idx / 252, c = idx - r * 252;
      xn[r][c] = (r < C_ && c < F_) ? x[((size_t)b * C_ + r) * F_ + c] : 0.f;
    }
    __syncthreads();
    if (tid < C_) {
      float s = 0.f;
      for (int k = 0; k < F_; ++k) s += xn[tid][k];
      float m = s / (float)F_;
      float v = 0.f;
      for (int k = 0; k < F_; ++k) { float d = xn[tid][k] - m; v += d * d; }
      float sd = sqrtf(v / (float)(F_ - 1));      // ddof=1
      mu[tid] = m;
      rs[tid] = 1.f / (sd + 1e-8f);
    } else if (tid == C_) { mu[C_] = 0.f; rs[C_] = 0.f; }
    __syncthreads();
    for (int idx = tid; idx < 64 * 252; idx += 512) {
      int r = idx / 252, c = idx - r * 252;
      xn[r][c] = (c < F_) ? (xn[r][c] - mu[r]) * rs[r] : 0.f;
    }
    __syncthreads();
    for (int k = 0; k < 252; k += 4) {
      v2f a, bv;
      a.x  = xn[ti * 16 + m16][k + khi + 0];
      a.y  = xn[ti * 16 + m16][k + khi + 1];
      bv.x = xn[tj * 16 + m16][k + khi + 0];
      bv.y = xn[tj * 16 + m16][k + khi + 1];
      acc = wmma4(a, bv, acc);
    }
    __syncthreads();
  }
  float* outp = padj + (size_t)blockIdx.x * 4096;
  for (int r = 0; r < 8; ++r) {
    int gm = ti * 16 + r + ((lane >> 4) << 3);
    int gn = tj * 16 + m16;
    outp[gm * 64 + gn] = acc[r];
  }
}

// Deterministic reduction of 64 partial grams -> adj (stored 64x64).
__global__ __launch_bounds__(256) void k_adjred(const float* __restrict__ padj,
                                                float* __restrict__ adj) {
  int idx = blockIdx.x * 256 + threadIdx.x;   // 16 blocks * 256 = 4096
  float s = 0.f;
  for (int b = 0; b < 64; ++b) s += padj[(size_t)b * 4096 + idx];
  adj[idx] = s;
}

// ---------------------------------------------------------------------------
// Kernel 2: per-row 8th-largest threshold (multiset semantics, matches
// top_k values[..., -1] with >= compare), build mask_full (63x63, 0/1).
// Scaling of adj by 1/(F*B) is monotonic and adj!=0 invariant, so skipped.
// ---------------------------------------------------------------------------
__global__ __launch_bounds__(64) void k_mask(const float* __restrict__ adj,
                                             float* __restrict__ maskf) {
  const int i = threadIdx.x;
  if (i >= C_) return;
  float thresh = 0.f;
  for (int j = 0; j < C_; ++j) {
    float v = adj[i * 64 + j];
    int gt = 0, ge = 0;
    for (int l = 0; l < C_; ++l) {
      float u = adj[i * 64 + l];
      gt += (u > v);
      ge += (u >= v);
    }
    if (gt < TOPK && ge >= TOPK) thresh = v;   // the 8th largest value
  }
  for (int j = 0; j < C_; ++j) {
    float a = adj[i * 64 + j];
    bool m = (i == j) || ((a >= thresh) && (a != 0.f));
    maskf[i * C_ + j] = m ? 1.f : 0.f;
  }
}

// ---------------------------------------------------------------------------
// Kernel 3: h0 = x[0] @ W_gat  (63x250 @ 250x1000), WMMA.
// 63 blocks over N tiles; 4 waves per block = 4 M tiles (rows padded to 64).
// ---------------------------------------------------------------------------
__global__ __launch_bounds__(128) void k_h0(const float* __restrict__ x,
                                            const float* __restrict__ Wg,
                                            float* __restrict__ h0) {
  const int tid = threadIdx.x, lane = tid & 31, wv = tid >> 5;
  const int m16 = lane & 15, khi = (lane >> 4) << 1;
  const int ncol = blockIdx.x * 16 + m16;
  const int row  = wv * 16 + m16;
  v8f acc = {};
  for (int kk = 0; kk < 252; kk += 4) {
    int k0 = kk + khi;
    v2f a, b;
    a.x = (row < C_ && k0 < F_)       ? x[(size_t)row * F_ + k0]       : 0.f;
    a.y = (row < C_ && k0 + 1 < F_)   ? x[(size_t)row * F_ + k0 + 1]   : 0.f;
    b.x = (k0 < F_ && ncol < HD_)     ? Wg[(size_t)k0 * HD_ + ncol]    : 0.f;
    b.y = (k0 + 1 < F_ && ncol < HD_) ? Wg[(size_t)(k0 + 1) * HD_ + ncol] : 0.f;
    acc = wmma4(a, b, acc);
  }
  for (int r = 0; r < 8; ++r) {
    int gm = wv * 16 + r + ((lane >> 4) << 3);
    int gn = blockIdx.x * 16 + m16;
    if (gm < C_ && gn < HD_) h0[(size_t)gm * HD_ + gn] = acc[r];
  }
}

// ---------------------------------------------------------------------------
// Kernel 4: batch-0 GAT attention. 63 blocks (one per destination node j).
// Tiny (63 nodes x 4 heads); VALU is fine here.
// ---------------------------------------------------------------------------
__global__ __launch_bounds__(256) void k_att(const float* __restrict__ h0,
                                             const float* __restrict__ asrc,
                                             const float* __restrict__ adst,
                                             const float* __restrict__ maskf,
                                             float* __restrict__ y0) {
  __shared__ float as_[C_ * H_];
  __shared__ float ad_[C_ * H_];
  __shared__ float wsm[C_ * H_];
  const int j = blockIdx.x, tid = threadIdx.x;
  for (int p = tid; p < C_ * H_; p += 256) {
    int i = p >> 2, h = p & 3;
    float s = 0.f, t = 0.f;
    const float* hr = h0 + (size_t)i * HD_ + h * D_;
    const float* sa = asrc + h * D_;
    const float* da = adst + h * D_;
    for (int d = 0; d < D_; ++d) { float hv = hr[d]; s += hv * sa[d]; t += hv * da[d]; }
    as_[p] = s; ad_[p] = t;
  }
  __syncthreads();
  if (tid < H_) {
    const int h = tid;
    const float adj_j = ad_[j * H_ + h];
    float mx = -3.4e38f;
    for (int i = 0; i < C_; ++i) {
      if (maskf[i * C_ + j] != 0.f) {
        float e = as_[i * H_ + h] + adj_j;
        e = (e >= 0.f) ? e : NEG * e;     // leaky_relu
        mx = fmaxf(mx, e);
      }
    }
    float sum = 0.f;
    for (int i = 0; i < C_; ++i) {
      float wv = 0.f;
      if (maskf[i * C_ + j] != 0.f) {
        float e = as_[i * H_ + h] + adj_j;
        e = (e >= 0.f) ? e : NEG * e;
        wv = expf(e - mx);
        sum += wv;
      }
      wsm[i * H_ + h] = wv;
    }
    float inv = 1.f / sum;                // diagonal always masked-in -> sum>0
    for (int i = 0; i < C_; ++i) wsm[i * H_ + h] *= inv;
  }
  __syncthreads();
  for (int p = tid; p < HD_; p += 256) {
    int h = p / D_;
    float s = 0.f;
    for (int i = 0; i < C_; ++i) s += wsm[i * H_ + h] * h0[(size_t)i * HD_ + p];
    y0[(size_t)j * HD_ + p] = s;          // att0 without bias
  }
}

// ---------------------------------------------------------------------------
// Kernel 5: fused out = ((x@W_gat [rows<63 -> att0]) + bias_gat) @ W_proj + b
// Per block: 16 output rows. Stage A (waves 0-3) computes a 16x64 K-chunk of
// the 1000-wide intermediate into LDS via WMMA; batch-0 rows overwritten with
// y0; Stage B (all 16 waves) accumulates the 16x256 (250) output tile via
// WMMA. The 258 MB intermediate never touches HBM.
// ---------------------------------------------------------------------------
__global__ __launch_bounds__(512) void k_fused(const float* __restrict__ x,
                                               const float* __restrict__ Wg,
                                               const float* __restrict__ bg,
                                               const float* __restrict__ Wp,
                                               const float* __restrict__ bp,
                                               const float* __restrict__ y0,
                                               float* __restrict__ out) {
  __shared__ float Xs[16][256];   // 16 rows x K(250->256)
  __shared__ float Ts[16][64];    // intermediate K-chunk
  const int tid = threadIdx.x, lane = tid & 31, wv = tid >> 5;
  const int m16 = lane & 15;
  const int khi = (lane >> 4) << 1;
  const int mhi = (lane >> 4) << 3;
  const int row0 = blockIdx.x * 16;

  for (int idx = tid; idx < 16 * 256; idx += 512) {
    int m = idx >> 8, k = idx & 255;
    Xs[m][k] = (k < F_) ? x[(size_t)(row0 + m) * F_ + k] : 0.f;
  }

  v8f oacc = {};
  for (int kc = 0; kc < 1024; kc += 64) {   // 16 chunks cover 1000 (padded)
    __syncthreads();                        // Xs ready / prev stage B done
    if (wv < 4) {                           // stage A: 16x64 chunk of x@Wg
      v8f t = {};
      const int n = kc + wv * 16 + m16;
      for (int kk = 0; kk < 252; kk += 4) {
        int k0 = kk + khi;
        v2f a, b;
        a.x = Xs[m16][k0];
        a.y = Xs[m16][k0 + 1];
        b.x = (k0 < F_ && n < HD_)     ? Wg[(size_t)k0 * HD_ + n]       : 0.f;
        b.y = (k0 + 1 < F_ && n < HD_) ? Wg[(size_t)(k0 + 1) * HD_ + n] : 0.f;
        t = wmma4(a, b, t);
      }
      for (int r = 0; r < 8; ++r) {
        int gk = kc + wv * 16 + m16;
        Ts[r + mhi][wv * 16 + m16] = (gk < HD_) ? (t[r] + bg[gk]) : 0.f;
      }
    }
    __syncthreads();
    if (row0 < C_) {                        // batch-0 rows -> attention output
      for (int idx = tid; idx < 16 * 64; idx += 512) {
        int m = idx >> 6, c = idx & 63;
        int r = row0 + m, gk = kc + c;
        if (r < C_ && gk < HD_) Ts[m][c] = y0[(size_t)r * HD_ + gk] + bg[gk];
      }
    }
    __syncthreads();
    {                                       // stage B: accumulate out tile
      const int n = wv * 16 + m16;
      for (int kk = 0; kk < 64; kk += 4) {
        int k0 = kk + khi;
        int gk = kc + k0;
        v2f a, b;
        a.x = Ts[m16][k0];
        a.y = Ts[m16][k0 + 1];
        b.x = (gk < HD_ && n < D_)     ? Wp[(size_t)gk * D_ + n]       : 0.f;
        b.y = (gk + 1 < HD_ && n < D_) ? Wp[(size_t)(gk + 1) * D_ + n] : 0.f;
        oacc = wmma4(a, b, oacc);
      }
    }
  }
  const int n = wv * 16 + m16;
  if (n < D_) {
    for (int r = 0; r < 8; ++r) {
      int gm = row0 + r + mhi;
      out[(size_t)gm * D_ + n] = oacc[r] + bp[n];
    }
  }
}

// ---------------------------------------------------------------------------
// Launch. Workspace layout (floats):
//   padj : [0,          262144)   64 blocks x 64x64 partial grams
//   adj  : [262144,     266240)   64x64
//   mask : [266240,     270336)   63x63 (0/1)
//   h0   : [270336,     333344)   63x1000
//   y0   : [333344,     396352)   63x1000   -> ~1.59 MB total
// ---------------------------------------------------------------------------
extern "C" void kernel_launch(void* const* d_in, const int* in_sizes, int n_in,
                              void* d_out, int out_size, void* d_ws, size_t ws_size,
                              hipStream_t stream) {
  const float* x    = (const float*)d_in[0];
  const float* Wg   = (const float*)d_in[1];
  const float* asrc = (const float*)d_in[2];
  const float* adst = (const float*)d_in[3];
  const float* bg   = (const float*)d_in[4];
  const float* Wp   = (const float*)d_in[5];
  const float* bp   = (const float*)d_in[6];
  float* out = (float*)d_out;

  float* w    = (float*)d_ws;
  float* padj = w;
  float* adj  = w + 262144;
  float* mskf = w + 266240;
  float* h0   = w + 270336;
  float* y0   = w + 333344;

  k_adj   <<<64,   512, 0, stream>>>(x, padj);
  k_adjred<<<16,   256, 0, stream>>>(padj, adj);
  k_mask  <<<1,     64, 0, stream>>>(adj, mskf);
  k_h0    <<<63,   128, 0, stream>>>(x, Wg, h0);
  k_att   <<<63,   256, 0, stream>>>(h0, asrc, adst, mskf, y0);
  k_fused <<<4032, 512, 0, stream>>>(x, Wg, bg, Wp, bp, y0, out);
}